// MVGeometricBilinear_30313879175205
// MI455X (gfx1250) — compile-verified
//
#include <hip/hip_runtime.h>
#include <utility>
#include <stdint.h>

typedef float v4f __attribute__((ext_vector_type(4)));

// ======================= constexpr G(3,0,1) blade tables =======================
// Blade index order matches the Python reference (itertools.combinations order),
// encoded as 4-bit masks (bit b = basis vector e_b present). e0^2 = 0, e1..e3^2 = 1.
namespace pga {

constexpr int MASKS[16] = {0x0, 0x1, 0x2, 0x4, 0x8, 0x3, 0x5, 0x9,
                           0x6, 0xA, 0xC, 0x7, 0xB, 0xD, 0xE, 0xF};

constexpr int idxOf(int m) {
  int r = -1;
  for (int i = 0; i < 16; ++i)
    if (MASKS[i] == m) r = i;
  return r;
}
constexpr int popc4(int v) { return (v & 1) + ((v >> 1) & 1) + ((v >> 2) & 1) + ((v >> 3) & 1); }
// Sign of e_A * e_B via bubble-count: for each element j of B, count elements of A > j.
constexpr int permSign(int a, int b) {
  int t = 0;
  for (int j = 0; j < 4; ++j)
    if ((b >> j) & 1) t += popc4(a >> (j + 1));
  return (t & 1) ? -1 : 1;
}
// Geometric product: zero iff e0 shared (degenerate metric), else permutation sign.
constexpr int gpSign(int i, int j) {
  return (MASKS[i] & MASKS[j] & 1) ? 0 : permSign(MASKS[i], MASKS[j]);
}
constexpr int gpK(int i, int j) { return idxOf(MASKS[i] ^ MASKS[j]); }
// join = undual(dual(x) ^ dual(y)); nonzero iff mi|mj == 1111b, result mask mi&mj.
// Sign = s_i * s_j * s_k * permSign(comp_i, comp_j) with s_X = permSign(X, comp(X)).
constexpr int joinSign(int i, int j) {
  int mi = MASKS[i], mj = MASKS[j];
  if ((mi | mj) != 0xF) return 0;
  int cmi = 0xF & ~mi, cmj = 0xF & ~mj, mk = mi & mj, cmk = 0xF & ~mk;
  return permSign(mi, cmi) * permSign(mj, cmj) * permSign(mk, cmk) * permSign(cmi, cmj);
}
constexpr int joinK(int i, int j) { return idxOf(MASKS[i] & MASKS[j]); }

} // namespace pga

// ---- compile-time-unrolled sparse bilinear contraction (only true terms emit) ----
template <int I, int J>
__device__ __forceinline__ void term(const float (&x)[16], const float (&y)[16],
                                     float (&gp)[16], float (&jn)[16]) {
  constexpr int gs = pga::gpSign(I, J);
  constexpr int js = pga::joinSign(I, J);
  if constexpr (gs != 0 || js != 0) {
    const float xv = x[I];
    const float yv = y[J];
    if constexpr (gs != 0) {
      constexpr int k = pga::gpK(I, J);
      if constexpr (gs > 0) gp[k] = __builtin_fmaf(xv, yv, gp[k]);
      else                  gp[k] = __builtin_fmaf(-xv, yv, gp[k]);
    }
    if constexpr (js != 0) {
      constexpr int k = pga::joinK(I, J);
      if constexpr (js > 0) jn[k] = __builtin_fmaf(xv, yv, jn[k]);
      else                  jn[k] = __builtin_fmaf(-xv, yv, jn[k]);
    }
  }
}

template <int I, int... Js>
__device__ __forceinline__ void termRow(std::integer_sequence<int, Js...>,
                                        const float (&x)[16], const float (&y)[16],
                                        float (&gp)[16], float (&jn)[16]) {
  (term<I, Js>(x, y, gp, jn), ...);
}
template <int... Is>
__device__ __forceinline__ void allTerms(std::integer_sequence<int, Is...>,
                                         const float (&x)[16], const float (&y)[16],
                                         float (&gp)[16], float (&jn)[16]) {
  (termRow<Is>(std::make_integer_sequence<int, 16>{}, x, y, gp, jn), ...);
}

__device__ __forceinline__ void compute_store(const float (&xr)[16], const float (&yr)[16],
                                              float r, float* __restrict__ op) {
  float gp[16] = {};
  float jn[16] = {};
  allTerms(std::make_integer_sequence<int, 16>{}, xr, yr, gp, jn);
#pragma unroll
  for (int q = 0; q < 4; ++q) {
    v4f g = {gp[4 * q + 0], gp[4 * q + 1], gp[4 * q + 2], gp[4 * q + 3]};
    __builtin_nontemporal_store(g, (v4f*)(op + 4 * q));
  }
#pragma unroll
  for (int q = 0; q < 4; ++q) {
    v4f j = {r * jn[4 * q + 0], r * jn[4 * q + 1], r * jn[4 * q + 2], r * jn[4 * q + 3]};
    __builtin_nontemporal_store(j, (v4f*)(op + 16 + 4 * q));
  }
}

// ======================= CDNA5 data movers =======================
#define TILE 128      // positions per block-tile (block = 128 threads = 4 waves)
#define XP 20         // LDS row pitch in floats (16 data + 4 pad; 80B keeps 16B align,
                      // removes ds_load_b128 bank conflicts)

typedef int      i32x4 __attribute__((__vector_size__(16)));
typedef int      i32x8 __attribute__((__vector_size__(32)));
typedef unsigned u32x4 __attribute__((__vector_size__(16)));
typedef __attribute__((address_space(1))) i32x4 gbl_i32x4;
typedef __attribute__((address_space(3))) i32x4 lds_i32x4;

#define HAVE_TDM __has_builtin(__builtin_amdgcn_tensor_load_to_lds)

#if HAVE_TDM
// One TDM descriptor moves a whole 2048-float (8KB) tile global->LDS and applies
// the 16-data/4-pad DWORD pitch natively (pad_interval=16 DW, pad_amount=4 DW).
// D# group0: [1:0]=count=1, [63:32]=lds_addr, [120:64]=global_addr, [127:126]=type 2.
// D# group1: [17:16]=data_size 4B, [20]=pad_enable, [24:22]=pad_interval(3 -> 16DW),
//            [31:25]=pad_amount(3 -> 4DW), tensor_dim0=tile_dim0=2048, dim1=1,
//            tensor_dim0_stride=2048.
__device__ __forceinline__ void tdm_load_2048f(const float* gsrc, unsigned lds_byte) {
  const unsigned long long ga = (unsigned long long)(uintptr_t)gsrc;
  u32x4 g0 = {1u, lds_byte, (unsigned)ga,
              ((unsigned)(ga >> 32) & 0x01FFFFFFu) | 0x80000000u};
  i32x8 g1 = {(int)((2u << 16) | (1u << 20) | (3u << 22) | (3u << 25)),
              (int)(2048u << 16),  // tensor_dim0[15:0] in bits [63:48]
              (int)(1u << 16),     // tensor_dim1 = 1 in bits [95:80]
              (int)(2048u << 16),  // tile_dim0 = 2048 in bits [127:112]
              1,                   // tile_dim1 = 1 in bits [143:128]
              2048,                // tensor_dim0_stride[31:0]
              0, 0};
  i32x4 z4 = {0, 0, 0, 0};
#if defined(__clang_major__) && (__clang_major__ >= 23)
  i32x8 z8 = {0, 0, 0, 0, 0, 0, 0, 0};
  __builtin_amdgcn_tensor_load_to_lds(g0, g1, z4, z4, z8, 0);
#else
  __builtin_amdgcn_tensor_load_to_lds(g0, g1, z4, z4, 0);
#endif
}

#if __has_builtin(__builtin_amdgcn_s_wait_tensorcnt)
#define WAIT_TENSOR(n) __builtin_amdgcn_s_wait_tensorcnt(n)
#else
#define WAIT_TENSOR(n) asm volatile("s_wait_tensorcnt " #n ::: "memory")
#endif

#else  // !HAVE_TDM: proven per-lane async global->LDS staging

__device__ __forceinline__ void async_copy_b128(void* lds_dst, const void* gsrc) {
#if __has_builtin(__builtin_amdgcn_global_load_async_to_lds_b128)
  __builtin_amdgcn_global_load_async_to_lds_b128(
      (gbl_i32x4*)gsrc, (lds_i32x4*)lds_dst, 0, 0);
#else
  asm volatile("global_load_async_to_lds_b128 %0, %1, off"
               :: "v"((unsigned)(uintptr_t)lds_dst), "v"(gsrc)
               : "memory");
#endif
}
#if __has_builtin(__builtin_amdgcn_s_wait_asynccnt)
#define WAIT_ASYNC(n) __builtin_amdgcn_s_wait_asynccnt(n)
#else
#define WAIT_ASYNC(n) asm volatile("s_wait_asynccnt " #n ::: "memory")
#endif

#endif  // HAVE_TDM

// ======================= kernels =======================
__device__ __forceinline__ void load16_lds(float (&d)[16], const float* s) {
#pragma unroll
  for (int c = 0; c < 4; ++c) {
    v4f v = *(const v4f*)(s + 4 * c);
    d[4 * c + 0] = v.x; d[4 * c + 1] = v.y; d[4 * c + 2] = v.z; d[4 * c + 3] = v.w;
  }
}

__global__ __launch_bounds__(TILE) void mvgeo_main(const float* __restrict__ xg,
                                                   const float* __restrict__ yg,
                                                   const float* __restrict__ refg,
                                                   float* __restrict__ outg,
                                                   int nTiles) {
  __shared__ __align__(16) float smem[2][2][TILE * XP];
  const int tid = threadIdx.x;
  int t = blockIdx.x;           // launcher guarantees gridDim.x <= nTiles
  int stage = 0;

#if HAVE_TDM
  // Wave0/lane0 drives the TDM (EXEC is ignored by tensor ops; per-wave TENSORcnt).
  auto issue = [&](int stg, int tile) {
    const long long base = (long long)tile * (TILE * 16);
    tdm_load_2048f(xg + base, (unsigned)(uintptr_t)&smem[stg][0][0]);
    tdm_load_2048f(yg + base, (unsigned)(uintptr_t)&smem[stg][1][0]);
  };
  if (tid == 0) issue(0, t);
  for (; t < nTiles; t += gridDim.x) {
    const long long nxt = (long long)t + gridDim.x;
    if (nxt < nTiles) {
      if (tid == 0) {
        issue(stage ^ 1, (int)nxt);  // prefetch next tile (2 more TDM ops in flight)
        WAIT_TENSOR(2);              // TDM in-order per wave: current tile's 2 are done
      }
    } else if (tid == 0) {
      WAIT_TENSOR(0);
    }
    __syncthreads();                 // publish TDM-written LDS to all waves
#else
  // Per tile, each thread stages its own row of x and y: 4+4 async b128 ops
  // => 8 outstanding ASYNCcnt per wave per tile.
  auto issue = [&](int stg, int tile) {
    const long long row = ((long long)tile * TILE + tid) * 16;
    const char* gx = (const char*)(xg + row);
    const char* gy = (const char*)(yg + row);
    char* lx = (char*)&smem[stg][0][tid * XP];
    char* ly = (char*)&smem[stg][1][tid * XP];
#pragma unroll
    for (int c = 0; c < 4; ++c) async_copy_b128(lx + c * 16, gx + c * 16);
#pragma unroll
    for (int c = 0; c < 4; ++c) async_copy_b128(ly + c * 16, gy + c * 16);
  };
  issue(0, t);
  for (; t < nTiles; t += gridDim.x) {
    const long long nxt = (long long)t + gridDim.x;
    if (nxt < nTiles) {
      issue(stage ^ 1, (int)nxt);
      WAIT_ASYNC(8);
    } else {
      WAIT_ASYNC(0);
    }
    __syncthreads();
#endif

    float xr[16], yr[16];
    load16_lds(xr, &smem[stage][0][tid * XP]);
    load16_lds(yr, &smem[stage][1][tid * XP]);

    const long long pos = (long long)t * TILE + tid;
    const float r = __builtin_nontemporal_load(refg + pos * 16 + 15);
    compute_store(xr, yr, r, outg + pos * 32);

    __syncthreads();               // buffer 'stage' reusable next iteration
    stage ^= 1;
  }
}

// Generic tail (direct global loads) for N not a multiple of TILE (unused for
// the reference shape, but keeps the launch deterministic and shape-agnostic).
__global__ void mvgeo_tail(const float* __restrict__ xg, const float* __restrict__ yg,
                           const float* __restrict__ refg, float* __restrict__ outg,
                           long long start, long long N) {
  long long pos = start + (long long)blockIdx.x * blockDim.x + threadIdx.x;
  if (pos >= N) return;
  float xr[16], yr[16];
#pragma unroll
  for (int c = 0; c < 4; ++c) {
    v4f vx = *(const v4f*)(xg + pos * 16 + 4 * c);
    v4f vy = *(const v4f*)(yg + pos * 16 + 4 * c);
    xr[4 * c + 0] = vx.x; xr[4 * c + 1] = vx.y; xr[4 * c + 2] = vx.z; xr[4 * c + 3] = vx.w;
    yr[4 * c + 0] = vy.x; yr[4 * c + 1] = vy.y; yr[4 * c + 2] = vy.z; yr[4 * c + 3] = vy.w;
  }
  const float r = refg[pos * 16 + 15];
  compute_store(xr, yr, r, outg + pos * 32);
}

extern "C" void kernel_launch(void* const* d_in, const int* in_sizes, int n_in,
                              void* d_out, int out_size, void* d_ws, size_t ws_size,
                              hipStream_t stream) {
  (void)n_in; (void)out_size; (void)d_ws; (void)ws_size;
  const float* x   = (const float*)d_in[0];
  const float* y   = (const float*)d_in[1];
  const float* ref = (const float*)d_in[2];
  float* out = (float*)d_out;

  const long long N = (long long)in_sizes[0] / 16;   // positions (B*T)
  const int nTiles = (int)(N / TILE);
  const long long rem = N % TILE;

  if (nTiles > 0) {
    int grid = nTiles < 2048 ? nTiles : 2048;        // each block loops >= 1 tile
    mvgeo_main<<<grid, TILE, 0, stream>>>(x, y, ref, out, nTiles);
  }
  if (rem > 0) {
    const long long start = N - rem;
    int blocks = (int)((rem + TILE - 1) / TILE);
    mvgeo_tail<<<blocks, TILE, 0, stream>>>(x, y, ref, out, start, N);
  }
}